// KNNGaussianBlur_48292612276670
// MI455X (gfx1250) — compile-verified
//
#include <hip/hip_runtime.h>

// Gaussian blur sigma=4, radius 12 (25 taps), C=3, H=W=4096, clamp-to-edge,
// with global-max normalize/denormalize. Fused separable blur: one workgroup
// produces a 64x64 tile; vertical + horizontal passes are banded-Toeplitz
// GEMMs on V_WMMA_F32_16X16X4_F32 (gfx1250, wave32).

#define IMG_H 4096
#define IMG_W 4096
#define IMG_C 3
#define KRAD  12
#define KTAPS 25

#define TILE     64                 // 64x64 output tile per workgroup
#define PADDED   (TILE + 2 * KRAD)  // 88
#define LDSTRIDE 92                 // 16B-aligned rows; conflict-free col walks

typedef __attribute__((ext_vector_type(2))) float v2f;
typedef __attribute__((ext_vector_type(8))) float v8f;

#if defined(__gfx1250__) && __has_builtin(__builtin_amdgcn_wmma_f32_16x16x4_f32)
#define USE_WMMA 1
#else
#define USE_WMMA 0
#endif

// ---------------------------------------------------------------- max reduce
__global__ __launch_bounds__(256) void ws_init_kernel(unsigned* ws) {
    if (threadIdx.x == 0 && blockIdx.x == 0) ws[0] = 0u;  // 0.0f bits
}

__global__ __launch_bounds__(256) void gmax_kernel(const float* __restrict__ in,
                                                   unsigned* __restrict__ ws,
                                                   int n4) {
    __shared__ float red[256];
    const float4* in4 = (const float4*)in;
    float m = 0.0f;
    for (size_t i = (size_t)blockIdx.x * 256 + threadIdx.x; i < (size_t)n4;
         i += (size_t)gridDim.x * 256) {
        float4 v = in4[i];
        m = fmaxf(m, fmaxf(fmaxf(v.x, v.y), fmaxf(v.z, v.w)));
    }
    red[threadIdx.x] = m;
    __syncthreads();
    for (int s = 128; s > 0; s >>= 1) {
        if ((int)threadIdx.x < s)
            red[threadIdx.x] = fmaxf(red[threadIdx.x], red[threadIdx.x + s]);
        __syncthreads();
    }
    // inputs are non-negative -> uint-bit compare == float compare
    if (threadIdx.x == 0) atomicMax(ws, __float_as_uint(red[0]));
}

// ---------------------------------------------------------------- fused blur
__global__ __launch_bounds__(256) void blur_fused_kernel(
    const float* __restrict__ in, float* __restrict__ out,
    const unsigned* __restrict__ wsmax) {
    __shared__ float sK[KTAPS];
    __shared__ float sIn[PADDED * LDSTRIDE];   // 88 x 92 (88 cols used)
    __shared__ float sTmp[TILE * LDSTRIDE];    // 64 x 92 (88 cols used)

    const int tid = threadIdx.x;
    const int ch  = blockIdx.z;
    const int r0  = blockIdx.y * TILE;
    const int c0  = blockIdx.x * TILE;

    const float mmax = __uint_as_float(*wsmax);
    const float inv  = 1.0f / mmax;

    // normalized Gaussian taps
    if (tid < KTAPS) {
        float x = (float)(tid - KRAD);
        sK[tid] = expf(-(x * x) * (1.0f / 32.0f));  // 2*sigma^2 = 32
    }
    __syncthreads();
    if (tid == 0) {
        float s = 0.0f;
        for (int i = 0; i < KTAPS; ++i) s += sK[i];
        float r = 1.0f / s;
        for (int i = 0; i < KTAPS; ++i) sK[i] *= r;
    }
    __syncthreads();

    // ---- cooperative halo load (normalize on the fly) ----
    const size_t base = (size_t)ch * IMG_H * IMG_W;
    const bool interior = (r0 >= KRAD) && (r0 + TILE + KRAD <= IMG_H) &&
                          (c0 >= KRAD) && (c0 + TILE + KRAD <= IMG_W);
    if (interior) {
        // rows are 16B aligned: (c0 - 12) % 4 == 0; LDSTRIDE % 4 == 0
        const float* src = in + base + (size_t)(r0 - KRAD) * IMG_W + (c0 - KRAD);
        for (int i = tid; i < PADDED * (PADDED / 4); i += 256) {
            int rr = i / (PADDED / 4);
            int c4 = i % (PADDED / 4);
            float4 v = *(const float4*)(src + (size_t)rr * IMG_W + c4 * 4);
            v.x *= inv; v.y *= inv; v.z *= inv; v.w *= inv;
            *(float4*)&sIn[rr * LDSTRIDE + c4 * 4] = v;
        }
    } else {
        for (int i = tid; i < PADDED * PADDED; i += 256) {
            int rr = i / PADDED, cc = i % PADDED;
            int gr = min(max(r0 + rr - KRAD, 0), IMG_H - 1);
            int gc = min(max(c0 + cc - KRAD, 0), IMG_W - 1);
            sIn[rr * LDSTRIDE + cc] = in[base + (size_t)gr * IMG_W + gc] * inv;
        }
    }
    __syncthreads();

#if USE_WMMA
    const int lane = tid & 31;
    const int wave = __builtin_amdgcn_readfirstlane(tid) >> 5;  // SGPR-uniform
    const int q    = lane & 15;   // row (A) / col (B/D) owned by this lane
    const int h    = lane >> 4;   // K-pair selector

    // Banded Toeplitz tap fragments, shared by both passes:
    // vertical A[m][s] = k[s-m] with m=q; horizontal B[s][n] = k[s-n] with n=q.
    v2f T[10];
#pragma unroll
    for (int j = 0; j < 10; ++j) {
        int s0 = 4 * j + 2 * h;
        int t0 = s0 - q, t1 = s0 + 1 - q;
        T[j].x = (t0 >= 0 && t0 < KTAPS) ? sK[t0] : 0.0f;
        T[j].y = (t1 >= 0 && t1 < KTAPS) ? sK[t1] : 0.0f;
    }

    // -------- vertical pass: sTmp[r][c] = sum_t k[t] * sIn[r+t][c]
    // 4 row-tiles x 6 col-tiles (cols 88..95 computed but not stored)
    for (int tile = wave; tile < 24; tile += 8) {   // scalar loop, full EXEC
        int rt = tile & 3;
        int ct = tile >> 2;
        int col  = ct * 16 + q;
        int colc = min(col, PADDED - 1);            // keep LDS reads in-bounds
        v8f acc = {0.f, 0.f, 0.f, 0.f, 0.f, 0.f, 0.f, 0.f};
#pragma unroll
        for (int j = 0; j < 10; ++j) {
            int r = rt * 16 + 4 * j + 2 * h;        // max 87
            v2f b;
            b.x = sIn[r * LDSTRIDE + colc];
            b.y = sIn[(r + 1) * LDSTRIDE + colc];
            acc = __builtin_amdgcn_wmma_f32_16x16x4_f32(
                false, T[j], false, b, (short)0, acc, false, false);
        }
        if (col < PADDED) {
#pragma unroll
            for (int v = 0; v < 8; ++v) {
                int rr = rt * 16 + v + 8 * h;       // D: VGPR v -> M = v + 8*h
                sTmp[rr * LDSTRIDE + col] = acc[v];
            }
        }
    }
    __syncthreads();

    // -------- horizontal pass: out[r][c] = sum_t k[t] * sTmp[r][c+t]
    for (int tile = wave; tile < 16; tile += 8) {   // scalar loop, full EXEC
        int rt = tile & 3;
        int ct = tile >> 2;
        v8f acc = {0.f, 0.f, 0.f, 0.f, 0.f, 0.f, 0.f, 0.f};
#pragma unroll
        for (int j = 0; j < 10; ++j) {
            int scol = ct * 16 + 4 * j + 2 * h;     // max 87
            v2f a;
            a.x = sTmp[(rt * 16 + q) * LDSTRIDE + scol];
            a.y = sTmp[(rt * 16 + q) * LDSTRIDE + scol + 1];
            acc = __builtin_amdgcn_wmma_f32_16x16x4_f32(
                false, a, false, T[j], (short)0, acc, false, false);
        }
#pragma unroll
        for (int v = 0; v < 8; ++v) {
            int row = r0 + rt * 16 + v + 8 * h;
            int colo = c0 + ct * 16 + q;
            out[base + (size_t)row * IMG_W + colo] = acc[v] * mmax;
        }
    }
#else
    // scalar fallback (also the host-pass parse path)
    for (int i = tid; i < TILE * PADDED; i += 256) {
        int r = i / PADDED, c = i % PADDED;
        float s = 0.0f;
#pragma unroll
        for (int t = 0; t < KTAPS; ++t) s += sK[t] * sIn[(r + t) * LDSTRIDE + c];
        sTmp[r * LDSTRIDE + c] = s;
    }
    __syncthreads();
    for (int i = tid; i < TILE * TILE; i += 256) {
        int r = i / TILE, c = i % TILE;
        float s = 0.0f;
#pragma unroll
        for (int t = 0; t < KTAPS; ++t) s += sK[t] * sTmp[r * LDSTRIDE + c + t];
        out[base + (size_t)(r0 + r) * IMG_W + (c0 + c)] = s * mmax;
    }
#endif
}

// ---------------------------------------------------------------- launch
extern "C" void kernel_launch(void* const* d_in, const int* in_sizes, int n_in,
                              void* d_out, int out_size, void* d_ws,
                              size_t ws_size, hipStream_t stream) {
    const float* img = (const float*)d_in[0];
    float* out = (float*)d_out;
    unsigned* wsm = (unsigned*)d_ws;
    const int n = in_sizes[0];  // 3*4096*4096

    ws_init_kernel<<<1, 256, 0, stream>>>(wsm);
    gmax_kernel<<<2048, 256, 0, stream>>>(img, wsm, n / 4);

    dim3 grid(IMG_W / TILE, IMG_H / TILE, IMG_C);
    blur_fused_kernel<<<grid, 256, 0, stream>>>(img, out, wsm);
}